// Voxelizer_13941463843130
// MI455X (gfx1250) — compile-verified
//
#include <hip/hip_runtime.h>

// ---- Voxelizer grid config (matches reference) ----
#define VOX_X_MIN  (-51.2f)
#define VOX_Y_MIN  (-51.2f)
#define VOX_Z_MIN  (-3.0f)
#define VOX_STEP   (0.2f)
#define VOX_ZSTEP  (0.2f)
#define VOX_W      512
#define VOX_H      512
#define VOX_D      40
#define VOX_T      5
#define VOX_N      100000

// Zero-fill the output with 16-byte stores (global_store_b128), grid-stride.
// 209.7 MB of stores: this is the bandwidth-dominant phase (~9 us @ 23.3 TB/s).
__global__ void vox_zero_kernel(float4* __restrict__ out4, int n4) {
    int stride = gridDim.x * blockDim.x;
    for (int i = blockIdx.x * blockDim.x + threadIdx.x; i < n4; i += stride) {
        out4[i] = make_float4(0.0f, 0.0f, 0.0f, 0.0f);
    }
}

// Scalar tail (in case out_size % 4 != 0; with this config it's 0 threads of work).
__global__ void vox_zero_tail_kernel(float* __restrict__ out, int begin, int n) {
    int i = begin + blockIdx.x * blockDim.x + threadIdx.x;
    if (i < n) out[i] = 0.0f;
}

// One lane per point of batch 0. float4 load -> floor-bin -> bounds check ->
// plain store of 1.0f (idempotent; duplicates race benignly, matching
// .set(1.0, mode='drop') semantics).
__global__ void vox_scatter_kernel(const float4* __restrict__ pts,
                                   float* __restrict__ out, int npts) {
    int p = blockIdx.x * blockDim.x + threadIdx.x;
    if (p >= npts) return;

    float4 v = pts[p];  // x, y, z, intensity (intensity unused)

    // Match reference numerics: floor((coord - MIN) / STEP), then int cast.
    int iw  = (int)floorf((v.x - VOX_X_MIN) / VOX_STEP);
    int ih  = (int)floorf((v.y - VOX_Y_MIN) / VOX_STEP);
    int idp = (int)floorf((v.z - VOX_Z_MIN) / VOX_ZSTEP);

    // Unsigned compare drops both negative and >=dim indices (mode='drop').
    if ((unsigned)iw < (unsigned)VOX_W &&
        (unsigned)ih < (unsigned)VOX_H &&
        (unsigned)idp < (unsigned)VOX_D) {
        int t = p / VOX_N;  // time index within batch 0
        size_t idx = ((((size_t)t * VOX_D + (size_t)idp) * VOX_H) + (size_t)ih) * VOX_W
                     + (size_t)iw;
        out[idx] = 1.0f;
    }
}

extern "C" void kernel_launch(void* const* d_in, const int* in_sizes, int n_in,
                              void* d_out, int out_size, void* d_ws, size_t ws_size,
                              hipStream_t stream) {
    (void)in_sizes; (void)n_in; (void)d_ws; (void)ws_size;

    const float* lidars = (const float*)d_in[0];  // [B=4, T=5, N=100000, 4] f32
    float* out = (float*)d_out;                   // [T*D=200, 512, 512] f32

    // ---- Phase 1: zero the whole output (store-bandwidth bound) ----
    int n4 = out_size / 4;           // float4 elements
    int tail_begin = n4 * 4;
    {
        const int threads = 256;     // 8 wave32 per block
        int total_threads_wanted = n4;
        int blocks = (total_threads_wanted + threads - 1) / threads;
        if (blocks > 65535 * 8) blocks = 65535 * 8;   // grid-stride covers rest
        if (blocks < 1) blocks = 1;
        vox_zero_kernel<<<blocks, threads, 0, stream>>>((float4*)out, n4);
    }
    if (tail_begin < out_size) {
        int tail = out_size - tail_begin;
        vox_zero_tail_kernel<<<(tail + 255) / 256, 256, 0, stream>>>(out, tail_begin,
                                                                     out_size);
    }

    // ---- Phase 2: scatter batch-0 points (stream-ordered after zero-fill) ----
    // Batch 0 is the first T*N points of the contiguous lidar tensor.
    const int npts = VOX_T * VOX_N;  // 500,000
    {
        const int threads = 256;
        int blocks = (npts + threads - 1) / threads;
        vox_scatter_kernel<<<blocks, threads, 0, stream>>>((const float4*)lidars,
                                                           out, npts);
    }
}